// PredictionCapsule_33861522162564
// MI455X (gfx1250) — compile-verified
//
#include <hip/hip_runtime.h>
#include <math.h>

typedef __attribute__((ext_vector_type(2))) float v2f;
typedef __attribute__((ext_vector_type(8))) float v8f;

#define B_SZ    32
#define N_CAPS  2048
#define D_INP   16
#define C_CAPS  64
#define D_OUTC  32
#define K_TOT   (N_CAPS * D_INP)      // 32768
#define J_TOT   (C_CAPS * D_OUTC)     // 2048
#define KGROUPS 8
#define WPB     8                     // waves per block
#define N_PER_WAVE (N_CAPS / (KGROUPS * WPB))  // 32 capsule rows per wave

// ---------------------------------------------------------------------------
// Stage 1: partial GEMM  S_part[kg, b, j] = sum_{k in chunk} x[b,k] * Wm[k,j]
//   Wm[k = n*16+i, j = c*32+o] = W[c,n,o,i]
// Grid: (128 column tiles of 16, 8 k-groups), 256 threads (8 waves).
// Each wave: M=32 x N=16 tile over K=512 via v_wmma_f32_16x16x4_f32.
// ---------------------------------------------------------------------------
__global__ __launch_bounds__(256)
void caps_gemm_partial(const float* __restrict__ x,
                       const float* __restrict__ W,
                       float* __restrict__ partial) {
  const int jtile = blockIdx.x;          // 0..127
  const int kg    = blockIdx.y;          // 0..7
  const int w     = threadIdx.x >> 5;    // wave in block, 0..7
  const int lane  = threadIdx.x & 31;
  const int lhalf = lane >> 4;           // 0: K=0,1   1: K=2,3
  const int l16   = lane & 15;

  const int jbase = jtile * 16;
  const int c     = jtile >> 1;          // output capsule
  const int obase = (jtile & 1) * 16;    // which half of D_OUT

  // B-fragment base: column j = jbase + l16  ->  o = obase + l16
  const float* pB = W + (size_t)c * (N_CAPS * D_OUTC * D_INP)
                      + (size_t)(obase + l16) * D_INP
                      + lhalf * 2;
  // A-fragment bases: rows b = l16 (tile 0) and b = 16 + l16 (tile 1)
  const float* pA0 = x + (size_t)l16 * K_TOT + lhalf * 2;
  const float* pA1 = pA0 + (size_t)16 * K_TOT;

  v8f acc0 = {};   // b = 0..15
  v8f acc1 = {};   // b = 16..31

  const int n0 = kg * (WPB * N_PER_WAVE) + w * N_PER_WAVE;

  for (int n = n0; n < n0 + N_PER_WAVE; ++n) {
    const float* wr  = pB  + (size_t)n * (D_OUTC * D_INP);
    const float* xr0 = pA0 + (size_t)n * D_INP;
    const float* xr1 = pA1 + (size_t)n * D_INP;
#pragma unroll
    for (int ii = 0; ii < D_INP; ii += 4) {
      v2f bf = *(const v2f*)(wr  + ii);
      v2f a0 = *(const v2f*)(xr0 + ii);
      v2f a1 = *(const v2f*)(xr1 + ii);
      // D = A(16x4,f32) * B(4x16,f32) + C ; 8 args: negA,A,negB,B,cmod,C,reuseA,reuseB
      acc0 = __builtin_amdgcn_wmma_f32_16x16x4_f32(
                 false, a0, false, bf, (short)0, acc0, false, false);
      acc1 = __builtin_amdgcn_wmma_f32_16x16x4_f32(
                 false, a1, false, bf, (short)0, acc1, false, false);
    }
  }

  // Reduce the 8 waves' 32x16 tiles in LDS, write one partial per workgroup.
  __shared__ float red[WPB][32][16];
  const int mlo = lhalf * 8;             // C-layout: vgpr r -> M = mlo + r
#pragma unroll
  for (int r = 0; r < 8; ++r) {
    red[w][mlo + r][l16]      = acc0[r];
    red[w][16 + mlo + r][l16] = acc1[r];
  }
  __syncthreads();

  const int t  = threadIdx.x;
  const int mm = t >> 4;                 // 0..15
  const int nn = t & 15;
  float s0 = 0.f, s1 = 0.f;
#pragma unroll
  for (int ww = 0; ww < WPB; ++ww) {
    s0 += red[ww][mm][nn];
    s1 += red[ww][mm + 16][nn];
  }
  partial[(size_t)(kg * B_SZ + mm)      * J_TOT + jbase + nn] = s0;
  partial[(size_t)(kg * B_SZ + mm + 16) * J_TOT + jbase + nn] = s1;
}

// ---------------------------------------------------------------------------
// Stage 2: deterministic reduction over k-groups + uniform routing (softmax of
// zeros = 1/64) + squash over D_OUT. One wave per (b,c); lane = o.
// ---------------------------------------------------------------------------
__global__ __launch_bounds__(256)
void caps_route_squash(const float* __restrict__ partial,
                       float* __restrict__ out) {
  const int w    = threadIdx.x >> 5;
  const int lane = threadIdx.x & 31;
  const int p    = blockIdx.x * 8 + w;   // (b,c) pair, 0..2047
  const int b    = p >> 6;
  const int c    = p & 63;
  const int j    = c * D_OUTC + lane;

  float s = 0.f;
#pragma unroll
  for (int kg = 0; kg < KGROUPS; ++kg)
    s += partial[(size_t)(kg * B_SZ + b) * J_TOT + j];
  s *= (1.0f / 64.0f);                   // c = softmax(0) over C=64

  float sq = s * s;
#pragma unroll
  for (int off = 16; off >= 1; off >>= 1)
    sq += __shfl_xor(sq, off, 32);       // ||s||^2 over D_OUT lanes

  const float scale = sq / ((1.0f + sq) * sqrtf(sq + 1e-7f));
  out[((size_t)b * C_CAPS + c) * D_OUTC + lane] = scale * s;
}

// ---------------------------------------------------------------------------
extern "C" void kernel_launch(void* const* d_in, const int* in_sizes, int n_in,
                              void* d_out, int out_size, void* d_ws, size_t ws_size,
                              hipStream_t stream) {
  (void)in_sizes; (void)n_in; (void)out_size; (void)ws_size;
  const float* x = (const float*)d_in[0];          // [32, 2048, 16]
  const float* W = (const float*)d_in[1];          // [64, 2048, 32, 16]
  float* out     = (float*)d_out;                  // [32, 64, 32]
  float* partial = (float*)d_ws;                   // KGROUPS*32*2048 floats = 2 MB

  caps_gemm_partial<<<dim3(J_TOT / 16, KGROUPS), 256, 0, stream>>>(x, W, partial);
  caps_route_squash<<<dim3((B_SZ * C_CAPS) / 8), 256, 0, stream>>>(partial, out);
}